// StableGATLayer_74543452389737
// MI455X (gfx1250) — compile-verified
//
#include <hip/hip_runtime.h>
#include <hip/hip_bf16.h>

typedef __attribute__((ext_vector_type(2))) float v2f;
typedef __attribute__((ext_vector_type(8))) float v8f;

#define XS_LD 66   // padded LDS row stride for x/h tile (floats)
#define WT_LD 65   // padded LDS row stride for transposed W (floats)

// ---------------------------------------------------------------------------
// Kernel 1: h = x @ W^T + b  (WMMA f32 16x16x4), plus per-node attention
// scores s_src[n,hd] = <h[n,hd,:], a[hd,0:8]>, s_dst[n,hd] = <h[n,hd,:], a[hd,8:16]>
// Block = 256 threads = 8 waves; each wave computes a 16-row tile of h.
// ---------------------------------------------------------------------------
__global__ __launch_bounds__(256) void gat_node_transform(
    const float* __restrict__ x, const float* __restrict__ W,
    const float* __restrict__ bias, const float* __restrict__ a,
    float* __restrict__ h, float* __restrict__ s_src, float* __restrict__ s_dst,
    int N)
{
    __shared__ float wT[64 * WT_LD];      // wT[k*WT_LD + j] = W[j*64 + k]
    __shared__ float xs[128 * XS_LD];     // x tile, later reused as h tile

    const int tid = threadIdx.x;
    const int rowBase = blockIdx.x * 128;

    // Load W transposed into LDS (coalesced global reads).
    for (int i = tid; i < 64 * 64; i += 256) {
        int j = i >> 6, k = i & 63;
        wT[k * WT_LD + j] = W[i];
    }
    // Stage 128 rows of x into LDS (coalesced; clamp OOB rows).
    for (int i = tid; i < 128 * 64; i += 256) {
        int r = i >> 6, c = i & 63;
        int gr = rowBase + r; if (gr > N - 1) gr = N - 1;
        xs[r * XS_LD + c] = x[(size_t)gr * 64 + c];
    }
    __syncthreads();

    const int wave = tid >> 5;
    const int lane = tid & 31;
    const int half = lane >> 4;     // 0: K pair (k0,k0+1), 1: (k0+2,k0+3)
    const int lan  = lane & 15;     // M index (A) / N index (B)

    float* xrow = &xs[wave * 16 * XS_LD];

    v8f acc[4];
    #pragma unroll
    for (int t = 0; t < 4; ++t) acc[t] = (v8f)(0.0f);

    #pragma unroll 4
    for (int k0 = 0; k0 < 64; k0 += 4) {
        const int ka = k0 + half * 2;
        v2f afrag;
        afrag.x = xrow[lan * XS_LD + ka];
        afrag.y = xrow[lan * XS_LD + ka + 1];
        #pragma unroll
        for (int t = 0; t < 4; ++t) {
            const int j = t * 16 + lan;
            v2f bfrag;
            bfrag.x = wT[ka * WT_LD + j];
            bfrag.y = wT[(ka + 1) * WT_LD + j];
            acc[t] = __builtin_amdgcn_wmma_f32_16x16x4_f32(
                false, afrag, false, bfrag, (short)0, acc[t], false, false);
        }
    }

    // Add bias and park the 16x64 h tile back into this wave's LDS region.
    // C layout: reg r, lane -> M = r + 8*half, N = t*16 + lan.
    #pragma unroll
    for (int t = 0; t < 4; ++t) {
        const float bj = bias[t * 16 + lan];
        #pragma unroll
        for (int r = 0; r < 8; ++r) {
            xrow[(r + 8 * half) * XS_LD + t * 16 + lan] = acc[t][r] + bj;
        }
    }
    __syncthreads();

    // Coalesced store of h to global.
    for (int i = tid; i < 128 * 64; i += 256) {
        int r = i >> 6, c = i & 63;
        int gr = rowBase + r;
        if (gr < N) h[(size_t)gr * 64 + c] = xs[r * XS_LD + c];
    }
    // Per-(node, head) attention scores: 128*8 = 1024 pairs.
    for (int i = tid; i < 1024; i += 256) {
        int r = i >> 3, hd = i & 7;
        int gr = rowBase + r;
        if (gr < N) {
            float s1 = 0.f, s2 = 0.f;
            #pragma unroll
            for (int d = 0; d < 8; ++d) {
                float v = xs[r * XS_LD + hd * 8 + d];
                s1 += v * a[hd * 16 + d];
                s2 += v * a[hd * 16 + 8 + d];
            }
            s_src[(size_t)gr * 8 + hd] = s1;
            s_dst[(size_t)gr * 8 + hd] = s2;
        }
    }
}

// ---------------------------------------------------------------------------
// Kernel 2: one wave32 per edge. LeakyReLU + softmax over 8 heads via
// shuffle, then out[src] += alpha[head] * h[dst] with native f32 atomics.
// ---------------------------------------------------------------------------
__global__ __launch_bounds__(256) void gat_edge_scatter(
    const int* __restrict__ ei, const float* __restrict__ h,
    const float* __restrict__ s_src, const float* __restrict__ s_dst,
    float* __restrict__ out, int E)
{
    const int wid = blockIdx.x * 8 + (threadIdx.x >> 5);
    if (wid >= E) return;                 // wave-uniform exit
    const int lane = threadIdx.x & 31;

    const int src = ei[wid];
    const int dst = ei[E + wid];

    // Every 8-lane group holds all 8 head logits (replicated) -> EXEC stays full.
    const int hd = lane & 7;
    float sc = s_src[(size_t)src * 8 + hd] + s_dst[(size_t)dst * 8 + hd];
    sc = sc >= 0.f ? sc : 0.2f * sc;      // LeakyReLU(0.2)

    // softmax over groups of 8 lanes
    float m = sc;
    m = fmaxf(m, __shfl_xor(m, 1, 8));
    m = fmaxf(m, __shfl_xor(m, 2, 8));
    m = fmaxf(m, __shfl_xor(m, 4, 8));
    float e = __expf(sc - m);
    float s = e;
    s += __shfl_xor(s, 1, 8);
    s += __shfl_xor(s, 2, 8);
    s += __shfl_xor(s, 4, 8);
    const float alpha = e / s;

    // lane handles elements 2*lane, 2*lane+1 -> head = lane>>2.
    // Lane (lane>>2) holds alpha for that head (its hd == lane>>2).
    const float av = __shfl(alpha, lane >> 2, 32);

    const int j0 = lane * 2;
    const float2 v = *(const float2*)(h + (size_t)dst * 64 + j0);
    float* op = out + (size_t)src * 64 + j0;
    unsafeAtomicAdd(op,     av * v.x);
    unsafeAtomicAdd(op + 1, av * v.y);
}

// ---------------------------------------------------------------------------
// Kernel 3: one wave32 per node. y = out + x; LayerNorm; L2 normalize.
// In-place on d_out (each element read once, then written once).
// ---------------------------------------------------------------------------
__global__ __launch_bounds__(256) void gat_finalize(
    float* __restrict__ out, const float* __restrict__ x,
    const float* __restrict__ lns, const float* __restrict__ lnb, int N)
{
    const int wid = blockIdx.x * 8 + (threadIdx.x >> 5);
    if (wid >= N) return;
    const int lane = threadIdx.x & 31;
    const int j0 = lane * 2;
    const size_t base = (size_t)wid * 64 + j0;

    const float2 acc = *(const float2*)(out + base);
    const float2 xv  = *(const float2*)(x + base);
    float y0 = acc.x + xv.x;
    float y1 = acc.y + xv.y;

    float p = y0 + y1;
    #pragma unroll
    for (int o = 16; o >= 1; o >>= 1) p += __shfl_xor(p, o, 32);
    const float mu = p * (1.0f / 64.0f);

    const float d0 = y0 - mu, d1 = y1 - mu;
    p = d0 * d0 + d1 * d1;
    #pragma unroll
    for (int o = 16; o >= 1; o >>= 1) p += __shfl_xor(p, o, 32);
    const float rstd = rsqrtf(p * (1.0f / 64.0f) + 1e-5f);

    const float z0 = d0 * rstd * lns[j0]     + lnb[j0];
    const float z1 = d1 * rstd * lns[j0 + 1] + lnb[j0 + 1];

    p = z0 * z0 + z1 * z1;
    #pragma unroll
    for (int o = 16; o >= 1; o >>= 1) p += __shfl_xor(p, o, 32);
    const float inv = 1.0f / fmaxf(sqrtf(p), 1e-12f);

    float2 r; r.x = z0 * inv; r.y = z1 * inv;
    *(float2*)(out + base) = r;
}

// ---------------------------------------------------------------------------
extern "C" void kernel_launch(void* const* d_in, const int* in_sizes, int n_in,
                              void* d_out, int out_size, void* d_ws, size_t ws_size,
                              hipStream_t stream) {
    const float* x   = (const float*)d_in[0];
    const int*   ei  = (const int*)  d_in[1];
    const float* W   = (const float*)d_in[2];
    const float* b   = (const float*)d_in[3];
    const float* a   = (const float*)d_in[4];
    const float* lns = (const float*)d_in[5];
    const float* lnb = (const float*)d_in[6];

    const int N = in_sizes[0] / 64;
    const int E = in_sizes[1] / 2;

    float* out = (float*)d_out;
    float* h     = (float*)d_ws;                 // N*64
    float* s_src = h + (size_t)N * 64;           // N*8
    float* s_dst = s_src + (size_t)N * 8;        // N*8

    // d_out doubles as the segment-sum accumulator.
    hipMemsetAsync(out, 0, (size_t)N * 64 * sizeof(float), stream);

    gat_node_transform<<<(N + 127) / 128, 256, 0, stream>>>(x, W, b, a, h, s_src, s_dst, N);
    gat_edge_scatter  <<<(E + 7) / 8,     256, 0, stream>>>(ei, h, s_src, s_dst, out, E);
    gat_finalize      <<<(N + 7) / 8,     256, 0, stream>>>(out, x, lns, lnb, N);
}